// FeatureNet_71021579207200
// MI455X (gfx1250) — compile-verified
//
#include <hip/hip_runtime.h>

// ---------------------------------------------------------------------------
// VoxelNet FeatureNet (VFE x2 + scatter) for gfx1250 (MI455X), wave32.
// GEMMs use V_WMMA_F32_16X16X4_F32 (exact f32 matrix math).
// Output tensor (720 MB) dominates bandwidth; intermediates are recomputed,
// never stored in HBM. BN stats via LDS + global float atomics.
// ---------------------------------------------------------------------------

typedef __attribute__((ext_vector_type(2))) float v2f;
typedef __attribute__((ext_vector_type(8))) float v8f;

#define T_PTS 35     // points per voxel
#define TP    48     // padded to 3 x 16 M-tiles
#define CIN   7
#define XS    8      // padded input K (7 -> 8)
#define C1    16
#define C2    32
#define DD    10
#define HH    400
#define WWW   352
#define EPSV  1e-5f
#define WPB   4      // waves (voxels) per block -> 128 threads

// ws layout (floats):
//  [0..15]  sum1   [16..31] sumsq1  [32..47] scale1  [48..63] shift1
//  [64..95] sum2   [96..127] sumsq2 [128..159] scale2 [160..191] shift2

__device__ __forceinline__ v8f wmma4(v2f a, v2f b, v8f c) {
  // D = A(16x4,f32) x B(4x16,f32) + C(16x16,f32)
  return __builtin_amdgcn_wmma_f32_16x16x4_f32(false, a, false, b,
                                               (short)0, c, false, false);
}

// ---- stage one voxel's features + validity mask into this wave's LDS ------
__device__ __forceinline__ void load_voxel(const float* __restrict__ fk,
                                           float* xbuf, float* maskv, int lane) {
  for (int i = lane; i < TP * XS; i += 32) xbuf[i] = 0.f;
  __syncthreads();
  for (int i = lane; i < T_PTS * CIN; i += 32) {
    int t = i / CIN, c = i - t * CIN;
    xbuf[t * XS + c] = fk[i];
  }
  __syncthreads();
  for (int t = lane; t < TP; t += 32) {
    float m = 0.f;
    if (t < T_PTS) {
      float mx = xbuf[t * XS + 0];
      #pragma unroll
      for (int c = 1; c < CIN; ++c) mx = fmaxf(mx, xbuf[t * XS + c]);
      m = (mx != 0.f) ? 1.f : 0.f;   // mask = (max over channels != 0)
    }
    maskv[t] = m;
  }
  __syncthreads();
}

// ---- GEMM1: [48x8(pad)] x [8(pad)x16] via 6 WMMAs; returns relu(.. + b1) --
// D-frag element (mt, r): row = mt*16 + r + 8*(lane>>4), channel = lane&15
__device__ __forceinline__ void gemm1_relu(const float* xbuf,
                                           const float* __restrict__ w1,
                                           const float* __restrict__ b1,
                                           int lane, float h1[3][8]) {
  const int n = lane & 15, hf = lane >> 4;
  v2f bf[2];
  #pragma unroll
  for (int kc = 0; kc < 2; ++kc) {           // B frag: V_r holds K = kc*4 + r + 2*half
    int k0 = kc * 4 + 2 * hf;
    bf[kc].x = (k0     < CIN) ? w1[k0 * C1 + n]       : 0.f;
    bf[kc].y = (k0 + 1 < CIN) ? w1[(k0 + 1) * C1 + n] : 0.f;
  }
  const v8f vzero = {};
  v8f d[3] = {vzero, vzero, vzero};
  #pragma unroll
  for (int kc = 0; kc < 2; ++kc) {
    #pragma unroll
    for (int mt = 0; mt < 3; ++mt) {
      const float* row = xbuf + (mt * 16 + n) * XS + kc * 4 + 2 * hf;  // A frag
      v2f a; a.x = row[0]; a.y = row[1];
      d[mt] = wmma4(a, bf[kc], d[mt]);
    }
  }
  float bias = b1[n];
  #pragma unroll
  for (int mt = 0; mt < 3; ++mt)
    #pragma unroll
    for (int r = 0; r < 8; ++r)
      h1[mt][r] = fmaxf(d[mt][r] + bias, 0.f);
}

// ---- BN1 + per-voxel max over T + concat, masked, into LDS cat[48][32] ----
__device__ __forceinline__ void build_cat(const float h1[3][8],
                                          const float* maskv,
                                          const float* __restrict__ scale1,
                                          const float* __restrict__ shift1,
                                          float* cat, int lane) {
  const int n = lane & 15, hf = lane >> 4;
  float sc = scale1[n], sh = shift1[n];
  float hbn[3][8];
  float aggp = -3.4e38f;
  #pragma unroll
  for (int mt = 0; mt < 3; ++mt)
    #pragma unroll
    for (int r = 0; r < 8; ++r) {
      int row = mt * 16 + r + 8 * hf;
      float v = h1[mt][r] * sc + sh;
      hbn[mt][r] = v;
      if (row < T_PTS) aggp = fmaxf(aggp, v);      // agg over real rows (pre-mask)
    }
  float agg = fmaxf(aggp, __shfl_xor(aggp, 16, 32));  // lanes l, l^16 share channel
  #pragma unroll
  for (int mt = 0; mt < 3; ++mt)
    #pragma unroll
    for (int r = 0; r < 8; ++r) {
      int row = mt * 16 + r + 8 * hf;
      float m = maskv[row];                        // 0 for pad rows
      cat[row * C2 + n]      = hbn[mt][r] * m;
      cat[row * C2 + 16 + n] = agg * m;
    }
  __syncthreads();
}

// ---- GEMM2: [48x32] x [32x32] via 48 WMMAs --------------------------------
__device__ __forceinline__ void gemm2_frags(const float* cat,
                                            const float* __restrict__ w2,
                                            int lane, v8f d2[3][2]) {
  const int n = lane & 15, hf = lane >> 4;
  v2f bf[8][2];
  #pragma unroll
  for (int kc = 0; kc < 8; ++kc) {
    int k0 = kc * 4 + 2 * hf;
    #pragma unroll
    for (int nt = 0; nt < 2; ++nt) {
      bf[kc][nt].x = w2[k0 * C2 + nt * 16 + n];
      bf[kc][nt].y = w2[(k0 + 1) * C2 + nt * 16 + n];
    }
  }
  const v8f vzero = {};
  #pragma unroll
  for (int mt = 0; mt < 3; ++mt)
    #pragma unroll
    for (int nt = 0; nt < 2; ++nt) d2[mt][nt] = vzero;
  #pragma unroll
  for (int kc = 0; kc < 8; ++kc) {
    #pragma unroll
    for (int mt = 0; mt < 3; ++mt) {
      const float* row = cat + (mt * 16 + n) * C2 + kc * 4 + 2 * hf;
      v2f a; a.x = row[0]; a.y = row[1];
      #pragma unroll
      for (int nt = 0; nt < 2; ++nt)
        d2[mt][nt] = wmma4(a, bf[kc][nt], d2[mt][nt]);
    }
  }
}

// ===========================================================================
__global__ void init_ws_kernel(float* ws) {
  int i = threadIdx.x;
  if (i < 192) ws[i] = 0.f;
}

__global__ void stats1_kernel(const float* __restrict__ feat,
                              const float* __restrict__ w1,
                              const float* __restrict__ b1,
                              float* __restrict__ ws, int K) {
  __shared__ float xs[WPB][TP * XS];
  __shared__ float masks[WPB][TP];
  __shared__ float ssum[C1], ssq[C1];
  const int lane = threadIdx.x & 31, wid = threadIdx.x >> 5;
  int k = blockIdx.x * WPB + wid;
  bool active = (k < K);
  const float* fk = feat + (size_t)(active ? k : 0) * (T_PTS * CIN);
  if (threadIdx.x < C1) { ssum[threadIdx.x] = 0.f; ssq[threadIdx.x] = 0.f; }
  load_voxel(fk, xs[wid], masks[wid], lane);
  float h1[3][8];
  gemm1_relu(xs[wid], w1, b1, lane, h1);
  const int hf = lane >> 4;
  float s = 0.f, sq = 0.f;
  #pragma unroll
  for (int mt = 0; mt < 3; ++mt)
    #pragma unroll
    for (int r = 0; r < 8; ++r) {
      int row = mt * 16 + r + 8 * hf;
      if (row < T_PTS) { float v = h1[mt][r]; s += v; sq += v * v; }
    }
  if (!active) { s = 0.f; sq = 0.f; }
  atomicAdd(&ssum[lane & 15], s);
  atomicAdd(&ssq[lane & 15], sq);
  __syncthreads();
  if (threadIdx.x < C1) {
    atomicAdd(&ws[0 + threadIdx.x], ssum[threadIdx.x]);
    atomicAdd(&ws[16 + threadIdx.x], ssq[threadIdx.x]);
  }
}

// stage 0: BN1 (16 ch @ 0/16 -> 32/48); stage 1: BN2 (32 ch @ 64/96 -> 128/160)
__global__ void finalize_kernel(float* ws, const float* __restrict__ g1,
                                const float* __restrict__ be1,
                                const float* __restrict__ g2,
                                const float* __restrict__ be2, float invN) {
  int i = threadIdx.x;
  if (i < C1) {
    float mean = ws[i] * invN;
    float var  = ws[16 + i] * invN - mean * mean;
    float sc   = g1[i] * rsqrtf(var + EPSV);
    ws[32 + i] = sc;
    ws[48 + i] = be1[i] - mean * sc;
  } else if (i >= 64 && i < 64 + C2) {
    int c = i - 64;
    float mean = ws[64 + c] * invN;
    float var  = ws[96 + c] * invN - mean * mean;
    float sc   = g2[c] * rsqrtf(var + EPSV);
    ws[128 + c] = sc;
    ws[160 + c] = be2[c] - mean * sc;
  }
}

__global__ void stats2_kernel(const float* __restrict__ feat,
                              const float* __restrict__ w1,
                              const float* __restrict__ b1,
                              const float* __restrict__ w2,
                              const float* __restrict__ b2,
                              float* __restrict__ ws, int K) {
  __shared__ float xs[WPB][TP * XS];
  __shared__ float cats[WPB][TP * C2];
  __shared__ float masks[WPB][TP];
  __shared__ float ssum[C2], ssq[C2];
  const int lane = threadIdx.x & 31, wid = threadIdx.x >> 5;
  int k = blockIdx.x * WPB + wid;
  bool active = (k < K);
  const float* fk = feat + (size_t)(active ? k : 0) * (T_PTS * CIN);
  if (threadIdx.x < C2) { ssum[threadIdx.x] = 0.f; ssq[threadIdx.x] = 0.f; }
  load_voxel(fk, xs[wid], masks[wid], lane);
  float h1[3][8];
  gemm1_relu(xs[wid], w1, b1, lane, h1);
  build_cat(h1, masks[wid], ws + 32, ws + 48, cats[wid], lane);
  v8f d2[3][2];
  gemm2_frags(cats[wid], w2, lane, d2);
  const int n = lane & 15, hf = lane >> 4;
  float s[2] = {0.f, 0.f}, sq[2] = {0.f, 0.f};
  #pragma unroll
  for (int nt = 0; nt < 2; ++nt) {
    float bias = b2[nt * 16 + n];
    #pragma unroll
    for (int mt = 0; mt < 3; ++mt)
      #pragma unroll
      for (int r = 0; r < 8; ++r) {
        int row = mt * 16 + r + 8 * hf;
        if (row < T_PTS) {
          float v = fmaxf(d2[mt][nt][r] + bias, 0.f);
          s[nt] += v; sq[nt] += v * v;
        }
      }
  }
  if (!active) { s[0] = s[1] = sq[0] = sq[1] = 0.f; }
  #pragma unroll
  for (int nt = 0; nt < 2; ++nt) {
    atomicAdd(&ssum[nt * 16 + n], s[nt]);
    atomicAdd(&ssq[nt * 16 + n], sq[nt]);
  }
  __syncthreads();
  if (threadIdx.x < C2) {
    atomicAdd(&ws[64 + threadIdx.x], ssum[threadIdx.x]);
    atomicAdd(&ws[96 + threadIdx.x], ssq[threadIdx.x]);
  }
}

__global__ void zero_out_kernel(float* __restrict__ out, long long n) {
  long long n4 = n >> 2;
  float4* o4 = (float4*)out;
  long long i = (long long)blockIdx.x * blockDim.x + threadIdx.x;
  long long stride = (long long)gridDim.x * blockDim.x;
  float4 z; z.x = z.y = z.z = z.w = 0.f;
  for (long long j = i; j < n4; j += stride) o4[j] = z;
  long long tail = n4 << 2;
  if (i < (n - tail)) out[tail + i] = 0.f;
}

__global__ void final_kernel(const float* __restrict__ feat,
                             const float* __restrict__ w1,
                             const float* __restrict__ b1,
                             const float* __restrict__ w2,
                             const float* __restrict__ b2,
                             const int* __restrict__ coord,
                             const float* __restrict__ ws,
                             float* __restrict__ out, int K) {
  __shared__ float xs[WPB][TP * XS];
  __shared__ float cats[WPB][TP * C2];
  __shared__ float masks[WPB][TP];
  const int lane = threadIdx.x & 31, wid = threadIdx.x >> 5;
  int k = blockIdx.x * WPB + wid;
  bool active = (k < K);
  const float* fk = feat + (size_t)(active ? k : 0) * (T_PTS * CIN);
  load_voxel(fk, xs[wid], masks[wid], lane);
  float h1[3][8];
  gemm1_relu(xs[wid], w1, b1, lane, h1);
  build_cat(h1, masks[wid], ws + 32, ws + 48, cats[wid], lane);
  v8f d2[3][2];
  gemm2_frags(cats[wid], w2, lane, d2);
  const int n = lane & 15, hf = lane >> 4;
  const float* maskp = masks[wid];
  float vh[2], va[2];
  #pragma unroll
  for (int nt = 0; nt < 2; ++nt) {
    int c = nt * 16 + n;
    float bias = b2[c], sc = ws[128 + c], sh = ws[160 + c];
    float aggp = -3.4e38f, vhp = -3.4e38f;
    #pragma unroll
    for (int mt = 0; mt < 3; ++mt)
      #pragma unroll
      for (int r = 0; r < 8; ++r) {
        int row = mt * 16 + r + 8 * hf;
        if (row < T_PTS) {
          float v = fmaxf(d2[mt][nt][r] + bias, 0.f) * sc + sh;  // BN2(relu)
          aggp = fmaxf(aggp, v);            // agg2 (pre-mask, all rows)
          vhp  = fmaxf(vhp, v * maskp[row]);
        }
      }
    float agg = fmaxf(aggp, __shfl_xor(aggp, 16, 32));
    vh[nt] = fmaxf(vhp, __shfl_xor(vhp, 16, 32));
    float vap = -3.4e38f;
    #pragma unroll
    for (int mt = 0; mt < 3; ++mt)
      #pragma unroll
      for (int r = 0; r < 8; ++r) {
        int row = mt * 16 + r + 8 * hf;
        if (row < T_PTS) vap = fmaxf(vap, agg * maskp[row]);
      }
    va[nt] = fmaxf(vap, __shfl_xor(vap, 16, 32));
  }
  if (active && lane < 16) {
    int cb = coord[(size_t)k * 4 + 0];
    int cd = coord[(size_t)k * 4 + 1];
    int ch = coord[(size_t)k * 4 + 2];
    int cw = coord[(size_t)k * 4 + 3];
    long long base = ((((long long)cb * DD + cd) * HH + ch) * WWW + cw) * 64LL;
    atomicAdd(&out[base + n],      vh[0]);
    atomicAdd(&out[base + 16 + n], vh[1]);
    atomicAdd(&out[base + 32 + n], va[0]);
    atomicAdd(&out[base + 48 + n], va[1]);
  }
}

// ===========================================================================
extern "C" void kernel_launch(void* const* d_in, const int* in_sizes, int n_in,
                              void* d_out, int out_size, void* d_ws, size_t ws_size,
                              hipStream_t stream) {
  const float* feat = (const float*)d_in[0];
  const float* w1   = (const float*)d_in[1];
  const float* b1   = (const float*)d_in[2];
  const float* g1   = (const float*)d_in[3];
  const float* be1  = (const float*)d_in[4];
  const float* w2   = (const float*)d_in[5];
  const float* b2   = (const float*)d_in[6];
  const float* g2   = (const float*)d_in[7];
  const float* be2  = (const float*)d_in[8];
  const int*   crd  = (const int*)d_in[9];
  float* out = (float*)d_out;
  float* ws  = (float*)d_ws;

  int K = in_sizes[0] / (T_PTS * CIN);
  float invN = 1.0f / ((float)K * (float)T_PTS);
  int vox_blocks = (K + WPB - 1) / WPB;

  hipLaunchKernelGGL(init_ws_kernel, dim3(1), dim3(256), 0, stream, ws);
  hipLaunchKernelGGL(stats1_kernel, dim3(vox_blocks), dim3(WPB * 32), 0, stream,
                     feat, w1, b1, ws, K);
  hipLaunchKernelGGL(finalize_kernel, dim3(1), dim3(128), 0, stream,
                     ws, g1, be1, g2, be2, invN);
  hipLaunchKernelGGL(stats2_kernel, dim3(vox_blocks), dim3(WPB * 32), 0, stream,
                     feat, w1, b1, w2, b2, ws, K);
  hipLaunchKernelGGL(finalize_kernel, dim3(1), dim3(128), 0, stream,
                     ws, g1, be1, g2, be2, invN);
  hipLaunchKernelGGL(zero_out_kernel, dim3(8192), dim3(256), 0, stream,
                     out, (long long)out_size);
  hipLaunchKernelGGL(final_kernel, dim3(vox_blocks), dim3(WPB * 32), 0, stream,
                     feat, w1, b1, w2, b2, crd, ws, out, K);
}